// ItemRegressionModel_22694607192172
// MI455X (gfx1250) — compile-verified
//
#include <hip/hip_runtime.h>

// ---------------------------------------------------------------------------
// Item-KNN regression:  out[b] = bu + bi[t] + (1/16) * sum_k w[qtu_k,t]*(r[u,qtu_k]-bu-bi[qtu_k])
//
// Latency-bound gather kernel. One wave32 = 16 samples. The length-16 weighted
// reduction is done with 4 chained V_WMMA_F32_16X16X4_F32 (the only full-fp32
// WMMA on CDNA5): A = resid (16x4), B = w (4x16), diagonal of D = per-sample
// dot products. The ISA lane layouts of A and B coincide (lane l, m=l&15,
// h=l>>4 holds K=2h,2h+1 per chunk), so each lane's gathers feed both
// operands directly. Neighbor-index pairs are loaded as int2 (b64, 8B-aligned)
// and all index loads are issued before the dependent data gathers to
// maximize memory-level parallelism in the two-level gather chain.
// ---------------------------------------------------------------------------

typedef __attribute__((ext_vector_type(2))) float v2f;
typedef __attribute__((ext_vector_type(8))) float v8f;

#define N_USERS 4000
#define N_ITEMS 2000
#define K_NBR   16
#define N_BATCH 16384

#define WAVES_PER_BLOCK   8
#define SAMPLES_PER_WAVE  16
#define SAMPLES_PER_BLOCK (WAVES_PER_BLOCK * SAMPLES_PER_WAVE)   // 128

__global__ __launch_bounds__(WAVES_PER_BLOCK * 32)
void knn_regress_wmma(const int*   __restrict__ u,
                      const int*   __restrict__ t,
                      const float* __restrict__ rating,   // [USERS][ITEMS]
                      const int*   __restrict__ qtus,     // [USERS][ITEMS][K]
                      const float* __restrict__ weight,   // [ITEMS][ITEMS]
                      const float* __restrict__ b_user,   // [USERS]
                      const float* __restrict__ b_item,   // [ITEMS]
                      float*       __restrict__ out)      // [B]
{
    __shared__ float dtile[WAVES_PER_BLOCK * 256];   // one 16x16 D tile per wave

    const int tid  = threadIdx.x;
    const int wave = tid >> 5;
    const int lane = tid & 31;
    const int m    = lane & 15;    // sample-in-wave == matrix row/col
    const int h    = lane >> 4;    // lane half selects K sub-pair

    const int s   = blockIdx.x * SAMPLES_PER_BLOCK + wave * SAMPLES_PER_WAVE + m;
    const int u_s = u[s];
    const int t_s = t[s];
    const float bu  = b_user[u_s];
    const float bit = b_item[t_s];           // hoisted: issue with other gathers

    const int*   qrow = qtus + ((size_t)u_s * N_ITEMS + t_s) * K_NBR;
    const float* rrow = rating + (size_t)u_s * N_ITEMS;

    // Level-1 gathers: all four 8B-aligned index pairs up front (global_load_b64).
    int2 qp[4];
#pragma unroll
    for (int c = 0; c < 4; ++c)
        qp[c] = *(const int2*)(qrow + 4 * c + 2 * h);

    // Level-2 gathers + 4 chained f32 WMMAs over K chunks of 4.
    // A (16x4 f32): lane m, half h holds v0=K(2h), v1=K(2h+1)  -> resid
    // B ( 4x16 f32): lanes 0-15 rows K0..1, lanes 16-31 K2..3  -> w
    v8f acc = {};
#pragma unroll
    for (int c = 0; c < 4; ++c) {
        const int q0 = qp[c].x;
        const int q1 = qp[c].y;

        const float w0 = weight[(size_t)q0 * N_ITEMS + t_s];
        const float w1 = weight[(size_t)q1 * N_ITEMS + t_s];
        const float res0 = rrow[q0] - bu - b_item[q0];
        const float res1 = rrow[q1] - bu - b_item[q1];

        v2f a = { res0, res1 };   // A = resid
        v2f b = { w0,   w1   };   // B = w
        // (neg_a, A, neg_b, B, c_mod, C, reuse_a, reuse_b)
        acc = __builtin_amdgcn_wmma_f32_16x16x4_f32(false, a, false, b,
                                                    (short)0, acc, false, false);
    }

    // D layout: VGPR r -> row (r + 8h), col = m.  Spill tile to LDS, read diagonal.
    float* my = &dtile[wave * 256];
#pragma unroll
    for (int r = 0; r < 8; ++r)
        my[(r + 8 * h) * 16 + m] = acc[r];

    __syncthreads();

    if (lane < 16) {
        const float dot = my[lane * 17];   // D[lane][lane]
        out[s] = bu + bit + dot * (1.0f / 16.0f);
    }
}

extern "C" void kernel_launch(void* const* d_in, const int* in_sizes, int n_in,
                              void* d_out, int out_size, void* d_ws, size_t ws_size,
                              hipStream_t stream)
{
    const int*   u_  = (const int*)  d_in[0];
    const int*   t_  = (const int*)  d_in[1];
    const float* rm  = (const float*)d_in[2];
    const int*   qt  = (const int*)  d_in[3];
    const float* w_  = (const float*)d_in[4];
    const float* bu  = (const float*)d_in[5];
    const float* bi  = (const float*)d_in[6];
    float* out = (float*)d_out;

    dim3 grid(N_BATCH / SAMPLES_PER_BLOCK);   // 128 blocks, exact
    dim3 block(WAVES_PER_BLOCK * 32);         // 256 threads = 8 waves

    hipLaunchKernelGGL(knn_regress_wmma, grid, block, 0, stream,
                       u_, t_, rm, qt, w_, bu, bi, out);
}